// BOWFeatures_86517821215729
// MI455X (gfx1250) — compile-verified
//
#include <hip/hip_runtime.h>
#include <hip/hip_bf16.h>

// BOW one-hot features for MI455X (gfx1250).
// Pure write-bandwidth problem: 671 MB of fp32 output, ~29 us floor at 23.3 TB/s.
// Strategy: one 256-thread block (8 wave32 waves) per (b, m) token position.
// Each block writes one contiguous 40 KB row with non-temporal global_store_b128
// (stream >> 192 MB L2, never re-read), values built branch-free with v_cndmask.

#define BOW_NTYPES 2048
#define BOW_K      5        // 2*WINDOW + 1, WINDOW = 2
#define BOW_L      1024
#define BOW_ROW    (BOW_K * BOW_NTYPES)   // 10240 floats = 40 KB per row

// Native clang vector type: required by __builtin_nontemporal_store
// (HIP's float4 class is rejected), lowers to one global_store_b128.
typedef __attribute__((ext_vector_type(4))) float v4f;

__global__ __launch_bounds__(256) void BOWFeatures_86517821215729_kernel(
    const int* __restrict__ tokens,   // [B, L] int32
    float* __restrict__ out)          // [B, L, K*NTYPES] fp32
{
    const int bm  = blockIdx.x;           // flat (b, m) index, 0 .. B*L-1
    const int b   = bm >> 10;             // / L
    const int m   = bm & (BOW_L - 1);     // % L
    const int tid = threadIdx.x;          // 0..255

    // Window tokens: uniform across the block -> compiler scalarizes to s_load.
    // Slot k looks at position n = m - (k - WINDOW) = m + 2 - k.
    // Invalid (out-of-range) slots get sentinel -1, which matches no element.
    int hot[BOW_K];
#pragma unroll
    for (int k = 0; k < BOW_K; ++k) {
        const int n      = m + 2 - k;
        const bool valid = (n >= 0) && (n < BOW_L);
        const int nc     = valid ? n : 0;               // clamped safe address
        const int t      = tokens[b * BOW_L + nc];
        hot[k] = valid ? t : -1;
    }

    float* __restrict__ row = out + (size_t)bm * BOW_ROW;

    // Each slot k spans 2048 floats = two 1024-float halves; 256 threads cover
    // each half with one contiguous float4 per thread (wave-contiguous b128s).
#pragma unroll
    for (int k = 0; k < BOW_K; ++k) {
        const int h  = hot[k];                 // block-uniform hot column
        const int la = h - (tid << 2);         // position within first half chunk
        const int lb = la - 1024;              // position within second half chunk

        v4f va, vb;
        va.x = (la == 0) ? 1.0f : 0.0f;
        va.y = (la == 1) ? 1.0f : 0.0f;
        va.z = (la == 2) ? 1.0f : 0.0f;
        va.w = (la == 3) ? 1.0f : 0.0f;
        vb.x = (lb == 0) ? 1.0f : 0.0f;
        vb.y = (lb == 1) ? 1.0f : 0.0f;
        vb.z = (lb == 2) ? 1.0f : 0.0f;
        vb.w = (lb == 3) ? 1.0f : 0.0f;

        v4f* pa = (v4f*)(row + k * BOW_NTYPES +        (tid << 2));
        v4f* pb = (v4f*)(row + k * BOW_NTYPES + 1024 + (tid << 2));
        __builtin_nontemporal_store(va, pa);   // global_store_b128 th:NT
        __builtin_nontemporal_store(vb, pb);   // global_store_b128 th:NT
    }
}

extern "C" void kernel_launch(void* const* d_in, const int* in_sizes, int n_in,
                              void* d_out, int out_size, void* d_ws, size_t ws_size,
                              hipStream_t stream) {
    (void)n_in; (void)d_ws; (void)ws_size; (void)out_size;
    const int*  tokens = (const int*)d_in[0];
    float*      out    = (float*)d_out;
    const int   n_pos  = in_sizes[0];         // B * L = 16384 rows
    BOWFeatures_86517821215729_kernel<<<n_pos, 256, 0, stream>>>(tokens, out);
}